// GCNLayer_5669356831736
// MI455X (gfx1250) — compile-verified
//
#include <hip/hip_runtime.h>

typedef float v2f __attribute__((ext_vector_type(2)));
typedef float v8f __attribute__((ext_vector_type(8)));

#define B_ 8
#define N_ 256
#define D_ 128
#define K_ 128
#define HSTRIDE 132   // padded LDS row stride (floats): bank-conflict-free A-frag reads

// ---------------- Kernel 1: xi = x @ w_vi ; xj = x @ w_vj ----------------
__global__ __launch_bounds__(128) void proj_kernel(const float* __restrict__ x,
                                                   const float* __restrict__ w_vi,
                                                   const float* __restrict__ w_vj,
                                                   float* __restrict__ xi,
                                                   float* __restrict__ xj) {
    const int row = blockIdx.x;                 // 0..B*N-1
    const int k   = threadIdx.x;                // 0..127
    const float* w   = (blockIdx.y == 0) ? w_vi : w_vj;
    float*       out = (blockIdx.y == 0) ? xi   : xj;
    const float* xr = x + (size_t)row * D_;
    float acc = 0.f;
#pragma unroll 8
    for (int d = 0; d < D_; ++d) acc = fmaf(xr[d], w[d * K_ + k], acc);
    out[(size_t)row * K_ + k] = acc;
}

// ---------------- Kernel 2: fused H / new_alpha / AH / new_x ----------------
// One workgroup (8 waves) per (b,i). Wave w owns output columns [16w, 16w+16).
__global__ __launch_bounds__(256) void fused_kernel(const float* __restrict__ alpha,
                                                    const float* __restrict__ adj,
                                                    const float* __restrict__ bias_h,
                                                    const float* __restrict__ w_node,
                                                    const float* __restrict__ w_alpha,
                                                    const float* __restrict__ xi,
                                                    const float* __restrict__ xj,
                                                    float* __restrict__ new_x,
                                                    float* __restrict__ new_alpha) {
    __shared__ float Hl[16 * HSTRIDE];   // 16x128 H tile, padded
    __shared__ float xib[D_];            // xi row + bias
    __shared__ float adjl[N_];           // adj row
    __shared__ float AH[D_];             // adj-weighted row sum

    const int tid  = threadIdx.x;
    const int lane = tid & 31;
    const int wv   = tid >> 5;           // wave id == nt (output col tile)
    const int half = lane >> 4;          // 0 or 1 (WMMA half-wave)
    const int l16  = lane & 15;
    const int bi   = blockIdx.x;         // b*N + i
    const int b    = bi >> 8;

    // Preload this wave's B-fragments of w_alpha into registers.
    // Frag layout (mirrors fp32 A table): VGPR v, lane L -> K = 4kc + v + 2*(L/16), N = L%16.
    v2f Bf[32];
    {
        const float* wp = w_alpha + (wv * 16 + l16);
#pragma unroll
        for (int kc = 0; kc < 32; ++kc) {
            const int dd = 4 * kc + 2 * half;
            Bf[kc].x = wp[(size_t)dd * K_];
            Bf[kc].y = wp[(size_t)(dd + 1) * K_];
        }
    }

    if (tid < D_) {
        xib[tid] = xi[(size_t)bi * D_ + tid] + bias_h[tid];
        AH[tid]  = 0.f;
    }
    adjl[tid] = adj[(size_t)bi * N_ + tid];
    __syncthreads();

    const int m  = tid >> 4;             // H-tile row this thread fills (0..15)
    const int g  = tid & 15;             // column group
    const int d0 = g * 8;                // 8 columns per thread

    float partAH[8];
#pragma unroll
    for (int c = 0; c < 8; ++c) partAH[c] = 0.f;

    const float* alpha_bi = alpha + (size_t)bi * N_ * D_;
    const float* xj_b     = xj + (size_t)b * N_ * D_;
    float*       na_bi    = new_alpha + (size_t)bi * N_ * K_;

    for (int jt = 0; jt < 16; ++jt) {
        const int j0 = jt * 16;
        // ---- Phase A: build H tile in LDS (coalesced alpha read, used once) ----
        {
            const int j = j0 + m;
            const float* ap = alpha_bi + (size_t)j * D_ + d0;
            const float* xp = xj_b + (size_t)j * D_ + d0;
            if (jt < 15) __builtin_prefetch(ap + 16 * D_, 0, 1);  // next tile
            const float aw = adjl[j];
#pragma unroll
            for (int c = 0; c < 8; ++c) {
                float h = ap[c] + xp[c] + xib[d0 + c];
                h = h > 0.f ? h : 0.f;
                Hl[m * HSTRIDE + d0 + c] = h;
                partAH[c] = fmaf(aw, h, partAH[c]);
            }
        }
        __syncthreads();

        // ---- Phase B: acc(16x16) = H(16x128) @ w_alpha[:, 16wv:16wv+16) ----
        v8f acc;
#pragma unroll
        for (int r = 0; r < 8; ++r) acc[r] = 0.f;
#pragma unroll
        for (int kc = 0; kc < 32; ++kc) {
            const int dd = 4 * kc + 2 * half;
            v2f Af = *(const v2f*)&Hl[l16 * HSTRIDE + dd];   // ds_load_b64, conflict-free
            acc = __builtin_amdgcn_wmma_f32_16x16x4_f32(
                    /*neg_a=*/false, Af, /*neg_b=*/false, Bf[kc],
                    /*c_mod=*/(short)0, acc, /*reuse_a=*/false, /*reuse_b=*/false);
        }
        // relu + store new_alpha slice. D layout: VGPR r, lane L -> M=r+8*(L/16), N=L%16.
        {
            const int col = wv * 16 + l16;
#pragma unroll
            for (int r = 0; r < 8; ++r) {
                const int rowj = j0 + r + 8 * half;
                const float v = acc[r] > 0.f ? acc[r] : 0.f;
                na_bi[(size_t)rowj * K_ + col] = v;
            }
        }
        __syncthreads();   // H tile consumed; safe to overwrite next iteration
    }

    // ---- AH reduction across the 16 row-threads sharing each column group ----
#pragma unroll
    for (int c = 0; c < 8; ++c) atomicAdd(&AH[d0 + c], partAH[c]);
    __syncthreads();

    // ---- new_x = relu(AH @ w_node) ----
    if (tid < K_) {
        float s = 0.f;
#pragma unroll 8
        for (int d = 0; d < D_; ++d) s = fmaf(AH[d], w_node[d * K_ + tid], s);
        new_x[(size_t)bi * K_ + tid] = s > 0.f ? s : 0.f;
    }
}

// ---------------- launcher ----------------
extern "C" void kernel_launch(void* const* d_in, const int* in_sizes, int n_in,
                              void* d_out, int out_size, void* d_ws, size_t ws_size,
                              hipStream_t stream) {
    const float* x       = (const float*)d_in[0];
    const float* alpha   = (const float*)d_in[1];
    const float* adj     = (const float*)d_in[2];
    /* d_in[3] box_num: unused in the math */
    const float* w_vi    = (const float*)d_in[4];
    const float* w_vj    = (const float*)d_in[5];
    const float* bias_h  = (const float*)d_in[6];
    const float* w_node  = (const float*)d_in[7];
    const float* w_alpha = (const float*)d_in[8];

    float* xi = (float*)d_ws;                       // (B*N, D) = 1 MB
    float* xj = xi + (size_t)B_ * N_ * D_;          // (B*N, D) = 1 MB

    float* new_x     = (float*)d_out;               // (B,N,K) first
    float* new_alpha = new_x + (size_t)B_ * N_ * K_;// then (B,N,N,K)

    dim3 g1(B_ * N_, 2);
    proj_kernel<<<g1, 128, 0, stream>>>(x, w_vi, w_vj, xi, xj);
    fused_kernel<<<B_ * N_, 256, 0, stream>>>(alpha, adj, bias_h, w_node, w_alpha,
                                              xi, xj, new_x, new_alpha);
}